// GraphSAINT_90159953477911
// MI455X (gfx1250) — compile-verified
//
#include <hip/hip_runtime.h>

typedef float v2f __attribute__((ext_vector_type(2)));
typedef float v8f __attribute__((ext_vector_type(8)));

#define HID 64

// ---------------------------------------------------------------- zero fill
__global__ void zero_kernel(float* __restrict__ p, long long n) {
    long long i = (long long)blockIdx.x * blockDim.x + threadIdx.x;
    if (i < n) p[i] = 0.0f;
}

// ---------------------------------------------------------------- degree
__global__ void degree_kernel(const int* __restrict__ dst,
                              float* __restrict__ deg, int E) {
    int e = blockIdx.x * blockDim.x + threadIdx.x;
    if (e < E) unsafeAtomicAdd(&deg[dst[e]], 1.0f);
}

// ------------------------------------------------- weighted scatter-add msg
// one thread per (edge, 4-feature chunk); FV = F/4 (power of two)
template <int F>
__global__ void scatter_kernel(const float* __restrict__ x,
                               const int* __restrict__ src,
                               const int* __restrict__ dst,
                               const float* __restrict__ w,
                               float* __restrict__ sum, int E) {
    constexpr int FV = F / 4;
    int tid = blockIdx.x * blockDim.x + threadIdx.x;
    if (tid >= E * FV) return;
    int e  = tid / FV;
    int f4 = tid & (FV - 1);
    int s = src[e], d = dst[e];
    float wt = w[e];
    const float4 v = ((const float4*)(x + (size_t)s * F))[f4];
    float* sp = sum + (size_t)d * F + (size_t)f4 * 4;
    unsafeAtomicAdd(sp + 0, v.x * wt);
    unsafeAtomicAdd(sp + 1, v.y * wt);
    unsafeAtomicAdd(sp + 2, v.z * wt);
    unsafeAtomicAdd(sp + 3, v.w * wt);
}

// ------------------------------------------------------- fused SAGE dense
// out[16 nodes, 64] = relu([sum/deg | x] @ [Wl ; Wr] + b)
// block = 128 threads (4 waves); wave w owns output columns [16w, 16w+16)
// WMMA: V_WMMA_F32_16X16X4_F32, K = 2*FIN accumulated 4 at a time.
template <int FIN>
__global__ __launch_bounds__(128) void sage_layer_kernel(
    const float* __restrict__ x,    // [N, FIN]
    const float* __restrict__ sum,  // [N, FIN]
    const float* __restrict__ deg,  // [N]
    const float* __restrict__ Wl,   // [FIN, 64] row-major
    const float* __restrict__ Wr,   // [FIN, 64] row-major
    const float* __restrict__ bias, // [64]
    float* __restrict__ out,        // [N, 64]
    int N) {
    constexpr int K   = 2 * FIN;
    constexpr int LDK = K + 4;                  // pad to dodge LDS bank conflicts
    __shared__ float tile[16 * LDK];

    const int node0 = blockIdx.x * 16;
    const int tid   = threadIdx.x;

    // stage A = [agg | x] for 16 nodes into LDS (row-major, stride LDK)
    for (int idx = tid; idx < 16 * K; idx += 128) {
        int m = idx / K;                         // K is a power of two
        int k = idx - m * K;
        int node = node0 + m;
        if (node >= N) node = N - 1;             // clamp (tiles are full anyway)
        float v;
        if (k < FIN) {
            float d = deg[node];
            v = sum[(size_t)node * FIN + k] / fmaxf(d, 1.0f);
        } else {
            v = x[(size_t)node * FIN + (k - FIN)];
        }
        tile[m * LDK + k] = v;
    }
    __syncthreads();

    const int lane = tid & 31;
    const int kh   = lane >> 4;                  // half-wave: K-pair select
    const int l15  = lane & 15;                  // M for A, N for B/C/D
    const int n0   = (tid >> 5) * 16;            // this wave's column tile
    const int col  = n0 + l15;

    v8f c = {};
    for (int k = 0; k < K; k += 4) {
        const int ka = k + 2 * kh;               // this half-wave's K offset
        v2f a, b;
        a.x = tile[l15 * LDK + ka];
        a.y = tile[l15 * LDK + ka + 1];
        // B rows k..k+3 all come from the same weight matrix (k multiple of 4)
        const float* W = (k < FIN) ? Wl : Wr;
        const int kk   = (k < FIN) ? ka : (ka - FIN);
        b.x = W[kk * HID + col];
        b.y = W[(kk + 1) * HID + col];
        c = __builtin_amdgcn_wmma_f32_16x16x4_f32(
                false, a, false, b, (short)0, c, false, false);
    }

    // C/D layout: VGPR r -> M = r + 8*half ; lanes 0..15 are N = col
    const float bv = bias[col];
    for (int r = 0; r < 8; ++r) {
        int m = r + 8 * kh;
        int node = node0 + m;
        if (node < N)
            out[(size_t)node * HID + col] = fmaxf(c[r] + bv, 0.0f);
    }
}

// --------------------------------------------- classifier + log_softmax
// one wave per 16 nodes: [x1|x2|x3] (16x192) @ Wlin (192x40, padded to 48)
__global__ __launch_bounds__(32) void classifier_kernel(
    const float* __restrict__ x1, const float* __restrict__ x2,
    const float* __restrict__ x3,
    const float* __restrict__ Wlin,  // [192, 40] row-major
    const float* __restrict__ blin,  // [40]
    float* __restrict__ out,         // [N, 40]
    int N) {
    constexpr int K = 192, LDK = 196, NC = 40, NCP = 48;
    __shared__ float tile[16 * LDK];
    __shared__ float logits[16 * NCP];

    const int node0 = blockIdx.x * 16;
    const int tid   = threadIdx.x;               // 0..31

    for (int idx = tid; idx < 16 * K; idx += 32) {
        int m = idx / K;
        int k = idx - m * K;
        int node = node0 + m;
        if (node >= N) node = N - 1;
        float v;
        if (k < 64)        v = x1[(size_t)node * 64 + k];
        else if (k < 128)  v = x2[(size_t)node * 64 + (k - 64)];
        else               v = x3[(size_t)node * 64 + (k - 128)];
        tile[m * LDK + k] = v;
    }
    __syncthreads();

    const int kh  = tid >> 4;
    const int l15 = tid & 15;

    for (int t = 0; t < 3; ++t) {                // column tiles 0..47
        const int col = t * 16 + l15;
        const bool valid = (col < NC);
        v8f c = {};
        for (int k = 0; k < K; k += 4) {
            const int ka = k + 2 * kh;
            v2f a, b;
            a.x = tile[l15 * LDK + ka];
            a.y = tile[l15 * LDK + ka + 1];
            b.x = valid ? Wlin[ka * NC + col] : 0.0f;
            b.y = valid ? Wlin[(ka + 1) * NC + col] : 0.0f;
            c = __builtin_amdgcn_wmma_f32_16x16x4_f32(
                    false, a, false, b, (short)0, c, false, false);
        }
        if (valid) {
            const float bv = blin[col];
            for (int r = 0; r < 8; ++r)
                logits[(r + 8 * kh) * NCP + col] = c[r] + bv;
        }
    }
    __syncthreads();

    // per-row log_softmax over the 40 real columns; lanes 0..15 own a row each
    if (tid < 16) {
        const int node = node0 + tid;
        if (node < N) {
            const float* row = logits + tid * NCP;
            float mx = row[0];
            for (int j = 1; j < NC; ++j) mx = fmaxf(mx, row[j]);
            float s = 0.0f;
            for (int j = 0; j < NC; ++j) s += __expf(row[j] - mx);
            const float ls = mx + __logf(s);
            for (int j = 0; j < NC; ++j)
                out[(size_t)node * NC + j] = row[j] - ls;
        }
    }
}

// ---------------------------------------------------------------- launcher
extern "C" void kernel_launch(void* const* d_in, const int* in_sizes, int n_in,
                              void* d_out, int out_size, void* d_ws, size_t ws_size,
                              hipStream_t stream) {
    const float* x0   = (const float*)d_in[0];
    const int*   ei   = (const int*)d_in[1];
    const float* ew   = (const float*)d_in[2];
    const float* W1l  = (const float*)d_in[3];
    const float* W1r  = (const float*)d_in[4];
    const float* b1   = (const float*)d_in[5];
    const float* W2l  = (const float*)d_in[6];
    const float* W2r  = (const float*)d_in[7];
    const float* b2   = (const float*)d_in[8];
    const float* W3l  = (const float*)d_in[9];
    const float* W3r  = (const float*)d_in[10];
    const float* b3   = (const float*)d_in[11];
    const float* Wlin = (const float*)d_in[12];
    const float* blin = (const float*)d_in[13];
    float* out = (float*)d_out;

    const int N = in_sizes[0] / 128;
    const int E = in_sizes[2];
    const int* src = ei;
    const int* dst = ei + E;

    // workspace: deg[N] | sum[N*128] | x1[N*64] | x2[N*64] | x3[N*64]
    float* deg = (float*)d_ws;
    float* sum = deg + N;
    float* x1  = sum + (size_t)N * 128;
    float* x2  = x1 + (size_t)N * 64;
    float* x3  = x2 + (size_t)N * 64;

    const int tiles = (N + 15) / 16;

    // zero deg + sum(128) in one pass (contiguous)
    {
        long long nz = (long long)N * 129;
        zero_kernel<<<(unsigned)((nz + 255) / 256), 256, 0, stream>>>(deg, nz);
    }
    degree_kernel<<<(E + 255) / 256, 256, 0, stream>>>(dst, deg, E);

    // ---- layer 1 (F=128)
    {
        long long tot = (long long)E * 32;
        scatter_kernel<128><<<(unsigned)((tot + 255) / 256), 256, 0, stream>>>(
            x0, src, dst, ew, sum, E);
    }
    sage_layer_kernel<128><<<tiles, 128, 0, stream>>>(x0, sum, deg, W1l, W1r, b1, x1, N);

    // ---- layer 2 (F=64)
    {
        long long nz = (long long)N * 64;
        zero_kernel<<<(unsigned)((nz + 255) / 256), 256, 0, stream>>>(sum, nz);
        long long tot = (long long)E * 16;
        scatter_kernel<64><<<(unsigned)((tot + 255) / 256), 256, 0, stream>>>(
            x1, src, dst, ew, sum, E);
    }
    sage_layer_kernel<64><<<tiles, 128, 0, stream>>>(x1, sum, deg, W2l, W2r, b2, x2, N);

    // ---- layer 3 (F=64)
    {
        long long nz = (long long)N * 64;
        zero_kernel<<<(unsigned)((nz + 255) / 256), 256, 0, stream>>>(sum, nz);
        long long tot = (long long)E * 16;
        scatter_kernel<64><<<(unsigned)((tot + 255) / 256), 256, 0, stream>>>(
            x2, src, dst, ew, sum, E);
    }
    sage_layer_kernel<64><<<tiles, 128, 0, stream>>>(x2, sum, deg, W3l, W3r, b3, x3, N);

    // ---- classifier + log_softmax
    classifier_kernel<<<tiles, 32, 0, stream>>>(x1, x2, x3, Wlin, blin, out, N);
}